// RvNN_49916109914203
// MI455X (gfx1250) — compile-verified
//
#include <hip/hip_runtime.h>
#include <math.h>

#define HIDDEN   100
#define MPAD     112      // 7 * 16 (M and K padding)
#define VOCAB    50000
#define NWORDS   40
#define NCLASS   4
#define CHUNK    16       // one WMMA N-tile of chain nodes
#define MAXCHAIN 5001     // chain is strictly decreasing from <=4999 -> <=5000 entries
#define NTHREADS 128      // 4 waves (wave32)

typedef float v2f __attribute__((ext_vector_type(2)));
typedef float v8f __attribute__((ext_vector_type(8)));

__launch_bounds__(NTHREADS)
__global__ void rvnn_chain_kernel(
    const float* __restrict__ xval, const int* __restrict__ xidx,
    const int*  __restrict__ tree, const int* __restrict__ nump,
    const float* __restrict__ E,
    const float* __restrict__ Wz, const float* __restrict__ Uz, const float* __restrict__ bz,
    const float* __restrict__ Wr, const float* __restrict__ Ur, const float* __restrict__ br,
    const float* __restrict__ Wh, const float* __restrict__ Uh, const float* __restrict__ bh,
    const float* __restrict__ Wout, const float* __restrict__ bout,
    float* __restrict__ out)
{
    __shared__ int   chain[MAXCHAIN];
    __shared__ int   sChainLen;
    __shared__ float XEs[MPAD][CHUNK];       // xe columns for this chunk (K x N)
    __shared__ float WXs[3][MPAD][CHUNK];    // Wz@xe, Wr@xe, Wh@xe
    __shared__ float hbuf[MPAD];
    __shared__ float rhbuf[MPAD];
    __shared__ float hnew[MPAD];

    const int tid  = threadIdx.x;
    const int lane = tid & 31;
    const int wave = tid >> 5;

    // ---- Phase 0: ancestor chain of node (num_parent - 1) -----------------
    // Output observes only child_hs[num_parent-1]; only its ancestor path
    // (expected length ~ln(N)) can influence it.
    if (tid == 0) {
        int node = nump[0] - 1;
        int len  = 0;
        while (len < MAXCHAIN) {
            chain[len++] = node;
            if (node == 0) break;
            node = tree[2 * node];           // tree[i][0] = parent
        }
        sChainLen = len;
    }
    for (int i = tid; i < MPAD; i += NTHREADS) hbuf[i] = 0.0f;  // node_h starts at 0
    __syncthreads();

    const int L = sChainLen;
    const float* Wmat[3] = { Wz, Wr, Wh };

    // Process chain root-first: chain[L-1], chain[L-2], ..., chain[0].
    for (int base = L - 1; base >= 0; base -= CHUNK) {
        const int cn = (base + 1 < CHUNK) ? (base + 1) : CHUNK;

        // ---- Phase 1: gather xe[:, n] = E[:, idx[node]] @ val[node] ------
        {
            const int n = tid >> 3;                       // node slot 0..15
            const int node = (n < cn) ? chain[base - n] : -1;
            for (int h = (tid & 7); h < MPAD; h += 8) {
                float acc = 0.0f;
                if (node >= 0 && h < HIDDEN) {
                    const float* vrow = xval + node * NWORDS;
                    const int*   irow = xidx + node * NWORDS;
                    const float* Erow = E + h * VOCAB;
                    #pragma unroll 4
                    for (int w = 0; w < NWORDS; ++w)
                        acc = fmaf(vrow[w], Erow[irow[w]], acc);
                }
                XEs[h][n] = acc;                           // pad rows/cols = 0
            }
        }
        __syncthreads();

        // ---- Phase 2: fp32 WMMA GEMM  WXs[g] = Wmat[g](100x100,pad) @ XEs
        // 21 (matrix, M-tile) jobs over 4 waves; wave-uniform control flow
        // so EXEC is all-ones for every v_wmma.
        for (int t = wave; t < 21; t += 4) {
            const int g  = t / 7;
            const int mt = t % 7;
            const float* W = Wmat[g];
            const int m    = mt * 16 + (lane & 15);        // A row (M)
            const int half = lane >> 4;                    // K sub-phase
            const int n    = lane & 15;                    // B/D column (N)
            v8f acc = {};
            for (int k = 0; k < MPAD; k += 4) {
                const int ka = k + 2 * half;
                v2f a, b;
                a.x = (m < HIDDEN && ka     < HIDDEN) ? W[m * HIDDEN + ka]     : 0.0f;
                a.y = (m < HIDDEN && ka + 1 < HIDDEN) ? W[m * HIDDEN + ka + 1] : 0.0f;
                b.x = XEs[ka][n];
                b.y = XEs[ka + 1][n];
                acc = __builtin_amdgcn_wmma_f32_16x16x4_f32(
                    false, a, false, b, (short)0, acc, false, false);
            }
            const int mrow = mt * 16 + half * 8;
            #pragma unroll
            for (int i = 0; i < 8; ++i)
                WXs[g][mrow + i][n] = acc[i];
        }
        __syncthreads();

        // ---- Phase 3: sequential GRU recurrence over this chunk ----------
        for (int s = 0; s < cn; ++s) {
            const int i = tid;
            float z = 0.0f, hprev = 0.0f;
            if (i < HIDDEN) {
                float uz = 0.0f, ur = 0.0f;
                for (int j = 0; j < HIDDEN; ++j) {
                    const float hj = hbuf[j];
                    uz = fmaf(Uz[i * HIDDEN + j], hj, uz);
                    ur = fmaf(Ur[i * HIDDEN + j], hj, ur);
                }
                z = WXs[0][i][s] + uz + bz[i];
                z = fminf(fmaxf(z, 0.0f), 1.0f);           // hardtanh(0,1)
                float r = WXs[1][i][s] + ur + br[i];
                r = fminf(fmaxf(r, 0.0f), 1.0f);
                hprev = hbuf[i];
                rhbuf[i] = hprev * r;
            }
            __syncthreads();
            if (i < HIDDEN) {
                float uh = 0.0f;
                for (int j = 0; j < HIDDEN; ++j)
                    uh = fmaf(Uh[i * HIDDEN + j], rhbuf[j], uh);
                float c = tanhf(WXs[2][i][s] + uh + bh[i]);
                hnew[i] = (1.0f - z) * hprev + z * c;
            }
            __syncthreads();
            if (i < HIDDEN) hbuf[i] = hnew[i];
            __syncthreads();
        }
    }

    // ---- Phase 4: softmax(W_out @ h + b_out) ------------------------------
    if (tid == 0) {
        float logits[NCLASS];
        float mx = -3.0e38f;
        for (int c = 0; c < NCLASS; ++c) {
            float acc = bout[c];
            for (int j = 0; j < HIDDEN; ++j)
                acc = fmaf(Wout[c * HIDDEN + j], hbuf[j], acc);
            logits[c] = acc;
            mx = fmaxf(mx, acc);
        }
        float denom = 0.0f;
        for (int c = 0; c < NCLASS; ++c) { logits[c] = expf(logits[c] - mx); denom += logits[c]; }
        for (int c = 0; c < NCLASS; ++c) out[c] = logits[c] / denom;
    }
}

extern "C" void kernel_launch(void* const* d_in, const int* in_sizes, int n_in,
                              void* d_out, int out_size, void* d_ws, size_t ws_size,
                              hipStream_t stream) {
    (void)in_sizes; (void)n_in; (void)out_size; (void)d_ws; (void)ws_size;
    const float* xval = (const float*)d_in[0];
    const int*   xidx = (const int*)  d_in[1];
    const int*   tree = (const int*)  d_in[2];
    const int*   nump = (const int*)  d_in[3];
    const float* E    = (const float*)d_in[4];
    const float* Wz   = (const float*)d_in[5];
    const float* Uz   = (const float*)d_in[6];
    const float* bz   = (const float*)d_in[7];
    const float* Wr   = (const float*)d_in[8];
    const float* Ur   = (const float*)d_in[9];
    const float* br   = (const float*)d_in[10];
    const float* Wh   = (const float*)d_in[11];
    const float* Uh   = (const float*)d_in[12];
    const float* bh   = (const float*)d_in[13];
    const float* Wout = (const float*)d_in[14];
    const float* bout = (const float*)d_in[15];
    float* out = (float*)d_out;

    rvnn_chain_kernel<<<1, NTHREADS, 0, stream>>>(
        xval, xidx, tree, nump, E,
        Wz, Uz, bz, Wr, Ur, br, Wh, Uh, bh, Wout, bout, out);
}